// LanguageModelCriterion_54468775248230
// MI455X (gfx1250) — compile-verified
//
#include <hip/hip_runtime.h>
#include <hip/hip_bf16.h>
#include <math.h>

// Problem constants from the reference
#define BB 32
#define TT 128
#define VV 32000
#define LL 196
#define DD 768

#define N_REP (BB * TT)   // 4096 rows of logits
#define N_IMG (BB * LL)   // 6272 patch rows

typedef __attribute__((ext_vector_type(4))) float f4;
typedef __attribute__((ext_vector_type(2))) float v2f;
typedef __attribute__((ext_vector_type(8))) float v8f;

// ---------------------------------------------------------------------------
// Kernel 1: masked token cross-entropy per (b,t) row.
// One 256-thread block (8 waves) per row. One-pass online log-sum-exp over
// V=32000 logits with float4 (b128) non-temporal loads. Rows with mask==0 are
// skipped entirely (halves HBM traffic for ~0.5-density masks).
// ---------------------------------------------------------------------------
__global__ __launch_bounds__(256) void ce_rows_kernel(
    const float* __restrict__ logits,   // [N_REP, V]
    const int*   __restrict__ targets,  // [N_REP]
    const float* __restrict__ mask,     // [N_REP]
    float*       __restrict__ partial)  // [N_REP] out: masked CE per row
{
    const int row = blockIdx.x;
    const float mk = mask[row];
    if (mk == 0.0f) {
        if (threadIdx.x == 0) partial[row] = 0.0f;
        return;
    }

    const float* __restrict__ x = logits + (size_t)row * VV;
    const f4* __restrict__ x4 = (const f4*)x;

    float m = -INFINITY;
    float s = 0.0f;

    // V/4 = 8000 float4 per row, strided across 256 threads
    for (int i = threadIdx.x; i < VV / 4; i += 256) {
        f4 v = __builtin_nontemporal_load(&x4[i]);
        float lm = fmaxf(fmaxf(v.x, v.y), fmaxf(v.z, v.w));
        if (lm > m) {              // rare after warm-up: single rescale of s
            s *= __expf(m - lm);   // exp(-inf)=0 handles first iteration
            m = lm;
        }
        s += __expf(v.x - m) + __expf(v.y - m) +
             __expf(v.z - m) + __expf(v.w - m);
    }

    // wave32 reduction of (m, s) pairs
    #pragma unroll
    for (int off = 16; off > 0; off >>= 1) {
        float mo = __shfl_xor(m, off, 32);
        float so = __shfl_xor(s, off, 32);
        float mn = fmaxf(m, mo);
        s = s * __expf(m - mn) + so * __expf(mo - mn);
        m = mn;
    }

    // cross-wave (8 waves) reduction through LDS
    __shared__ float sm_m[8];
    __shared__ float sm_s[8];
    const int wave = threadIdx.x >> 5;
    const int lane = threadIdx.x & 31;
    if (lane == 0) { sm_m[wave] = m; sm_s[wave] = s; }
    __syncthreads();

    if (threadIdx.x == 0) {
        float M = sm_m[0], S = sm_s[0];
        #pragma unroll
        for (int w = 1; w < 8; ++w) {
            float mo = sm_m[w], so = sm_s[w];
            float mn = fmaxf(M, mo);
            S = S * __expf(M - mn) + so * __expf(mo - mn);
            M = mn;
        }
        const float xt = x[targets[row]];
        // ce = logsumexp - x_target
        partial[row] = (M + __logf(S) - xt) * mk;
    }
}

// ---------------------------------------------------------------------------
// Kernel 2: masked per-patch MSE. One wave per row (8 rows per 256-thread
// block). Each lane handles 6 float4 from pred and true. Masked rows skipped.
// ---------------------------------------------------------------------------
__global__ __launch_bounds__(256) void mse_rows_kernel(
    const float* __restrict__ t,       // images_true [N_IMG, D]
    const float* __restrict__ p,       // images_pred [N_IMG, D]
    const float* __restrict__ mask,    // [N_IMG]
    float*       __restrict__ partial) // [N_IMG] out: masked mean-sq per row
{
    const int wave = threadIdx.x >> 5;
    const int lane = threadIdx.x & 31;
    const int row  = blockIdx.x * 8 + wave;

    const float mk = mask[row];
    if (mk == 0.0f) {
        if (lane == 0) partial[row] = 0.0f;
        return;
    }

    const f4* __restrict__ t4 = (const f4*)(t + (size_t)row * DD);
    const f4* __restrict__ p4 = (const f4*)(p + (size_t)row * DD);

    float acc = 0.0f;
    #pragma unroll
    for (int i = lane; i < DD / 4; i += 32) {   // 192/32 = 6 iterations
        f4 a = __builtin_nontemporal_load(&t4[i]);
        f4 b = __builtin_nontemporal_load(&p4[i]);
        float dx = b.x - a.x, dy = b.y - a.y, dz = b.z - a.z, dw = b.w - a.w;
        acc += dx * dx + dy * dy + dz * dz + dw * dw;
    }

    #pragma unroll
    for (int off = 16; off > 0; off >>= 1)
        acc += __shfl_xor(acc, off, 32);

    if (lane == 0)
        partial[row] = acc * (1.0f / (float)DD) * mk;
}

// ---------------------------------------------------------------------------
// Kernel 3: single-wave WMMA reduction + final combine.
// Sums an array (length multiple of 64) with V_WMMA_F32_16X16X4_F32:
//   D = A(16x4) x ones(4x16) + C  => every column of D holds per-row sums.
// Each WMMA consumes 64 floats (2 A VGPRs). Column 0 of the f32 C/D layout
// lives in lane 0 (M=0..7, VGPR0..7) and lane 16 (M=8..15), so the grand
// total = colsum(lane0) + colsum(lane16). Exact f32 accumulation.
// Wave-uniform control flow keeps EXEC all-1s as WMMA requires.
// ---------------------------------------------------------------------------
__device__ inline float wmma_sum64(const float* __restrict__ d, int n, int lane)
{
    v2f ones; ones.x = 1.0f; ones.y = 1.0f;
    v8f acc = {};
    for (int base = 0; base < n; base += 64) {
        v2f a;
        a.x = d[base + lane];
        a.y = d[base + 32 + lane];
        acc = __builtin_amdgcn_wmma_f32_16x16x4_f32(
            /*neg_a=*/false, a, /*neg_b=*/false, ones,
            /*c_mod=*/(short)0, acc, /*reuse_a=*/false, /*reuse_b=*/false);
    }
    float colsum = acc[0] + acc[1] + acc[2] + acc[3] +
                   acc[4] + acc[5] + acc[6] + acc[7];
    colsum += __shfl_down(colsum, 16, 32);
    return colsum;   // valid in lane 0
}

__global__ __launch_bounds__(32) void reduce_combine_kernel(
    const float* __restrict__ rep_partial,  // [N_REP]
    const float* __restrict__ rep_mask,     // [N_REP]
    const float* __restrict__ img_partial,  // [N_IMG]
    const float* __restrict__ img_mask,     // [N_IMG]
    const float* __restrict__ rep_weight,   // scalar
    const float* __restrict__ img_weight,   // scalar
    float*       __restrict__ out)          // [1]
{
    const int lane = threadIdx.x;   // single wave of 32

    float ce_sum  = wmma_sum64(rep_partial, N_REP, lane);
    float rm_sum  = wmma_sum64(rep_mask,    N_REP, lane);
    float mse_sum = wmma_sum64(img_partial, N_IMG, lane);
    float im_sum  = wmma_sum64(img_mask,    N_IMG, lane);

    if (lane == 0) {
        float report_loss = ce_sum  / rm_sum * rep_weight[0];
        float image_loss  = mse_sum / im_sum * img_weight[0];
        out[0] = report_loss + image_loss;
    }
}

// ---------------------------------------------------------------------------
extern "C" void kernel_launch(void* const* d_in, const int* in_sizes, int n_in,
                              void* d_out, int out_size, void* d_ws, size_t ws_size,
                              hipStream_t stream) {
    const float* images_true    = (const float*)d_in[0];
    const int*   reports_true   = (const int*)  d_in[1];
    const float* images_pred    = (const float*)d_in[2];
    const float* reports_pred   = (const float*)d_in[3];
    const float* images_mask    = (const float*)d_in[4];
    const float* reports_mask   = (const float*)d_in[5];
    const float* images_weight  = (const float*)d_in[6];
    const float* reports_weight = (const float*)d_in[7];
    float* out = (float*)d_out;

    float* ws          = (float*)d_ws;
    float* rep_partial = ws;            // N_REP  floats
    float* img_partial = ws + N_REP;    // N_IMG  floats

    // 1) masked CE per logit row (dominant: ~512 MB stream, mask-skipped)
    ce_rows_kernel<<<N_REP, 256, 0, stream>>>(
        reports_pred, reports_true, reports_mask, rep_partial);

    // 2) masked MSE per patch row (~38.5 MB stream, mask-skipped)
    mse_rows_kernel<<<N_IMG / 8, 256, 0, stream>>>(
        images_true, images_pred, images_mask, img_partial);

    // 3) WMMA-based final reductions + scalar combine
    reduce_combine_kernel<<<1, 32, 0, stream>>>(
        rep_partial, reports_mask, img_partial, images_mask,
        reports_weight, images_weight, out);
}